// BincountMinlengthModule_38474317038177
// MI455X (gfx1250) — compile-verified
//
#include <hip/hip_runtime.h>

// Bincount(x, minlength=600) over 33,554,432 int32 values in [0, 600).
// Memory-bound: 128 MiB read @ 23.3 TB/s => ~5.7 us floor. Strategy:
//   1) zero 600-entry u32 accumulator in d_ws
//   2) grid-stride histogram: NT b128 loads -> LDS-privatized ds_add_u32
//      (8 replicated sub-histograms per block to cut wave32 same-bin
//      serialization), then one global_atomic_add_u32 flush per bin/block
//   3) finalize: convert u32 counts -> float into d_out

#define NBINS      600
#define NSUB       8            // LDS sub-histogram copies per block
#define SUBSTRIDE  600          // dwords per copy (600 % 64 = 24 -> bank de-alias)
#define LDS_WORDS  (NSUB * SUBSTRIDE)   // 4800 dwords = 19.2 KB of 320 KB WGP LDS

#define HIST_BLOCKS  1024
#define HIST_THREADS 256        // 8 wave32 waves per block

typedef int v4i __attribute__((ext_vector_type(4)));

__global__ __launch_bounds__(256)
void bincount_zero_ws(unsigned int* __restrict__ ws) {
    int i = blockIdx.x * blockDim.x + threadIdx.x;
    if (i < NBINS) ws[i] = 0u;
}

__global__ __launch_bounds__(HIST_THREADS)
void bincount_hist(const int* __restrict__ x, unsigned int* __restrict__ ws, int n) {
    __shared__ unsigned int lh[LDS_WORDS];

    // Clear LDS sub-histograms.
    for (int i = threadIdx.x; i < LDS_WORDS; i += HIST_THREADS) lh[i] = 0u;
    __syncthreads();

    const int n4     = n >> 2;                 // number of int4 packets
    const int tail   = n & 3;
    const unsigned sub = (threadIdx.x & (NSUB - 1)) * SUBSTRIDE;
    const int stride = gridDim.x * HIST_THREADS;
    const v4i* __restrict__ x4 = (const v4i*)x;

    for (int i = blockIdx.x * HIST_THREADS + threadIdx.x; i < n4; i += stride) {
        // Prefetch one grid-stride ahead (gfx1250 global_prefetch_b8).
        if (i + stride < n4)
            __builtin_prefetch(&x4[i + stride], 0, 1);
        // Streamed once: non-temporal 128-bit load, don't pollute L2.
        v4i v = __builtin_nontemporal_load(&x4[i]);
        atomicAdd(&lh[sub + (unsigned)v.x], 1u);   // ds_add_u32 (no return)
        atomicAdd(&lh[sub + (unsigned)v.y], 1u);
        atomicAdd(&lh[sub + (unsigned)v.z], 1u);
        atomicAdd(&lh[sub + (unsigned)v.w], 1u);
    }

    // Scalar tail (n % 4 != 0), handled by block 0 only.
    if (blockIdx.x == 0 && (int)threadIdx.x < tail) {
        int v = x[(n4 << 2) + threadIdx.x];
        atomicAdd(&lh[sub + (unsigned)v], 1u);
    }

    __syncthreads();

    // Reduce the NSUB copies and flush to the global accumulator.
    for (int b = threadIdx.x; b < NBINS; b += HIST_THREADS) {
        unsigned int s = 0u;
        #pragma unroll
        for (int c = 0; c < NSUB; ++c) s += lh[c * SUBSTRIDE + b];
        if (s) atomicAdd(&ws[b], s);               // global_atomic_add_u32
    }
}

__global__ __launch_bounds__(256)
void bincount_finalize(const unsigned int* __restrict__ ws, float* __restrict__ out) {
    int i = blockIdx.x * blockDim.x + threadIdx.x;
    if (i < NBINS) out[i] = (float)ws[i];          // counts < 2^24: exact in f32
}

extern "C" void kernel_launch(void* const* d_in, const int* in_sizes, int n_in,
                              void* d_out, int out_size, void* d_ws, size_t ws_size,
                              hipStream_t stream) {
    const int* x = (const int*)d_in[0];
    const int n  = in_sizes[0];
    unsigned int* ws = (unsigned int*)d_ws;        // 600 * 4 = 2400 bytes used
    float* out = (float*)d_out;

    bincount_zero_ws<<<(NBINS + 255) / 256, 256, 0, stream>>>(ws);
    bincount_hist<<<HIST_BLOCKS, HIST_THREADS, 0, stream>>>(x, ws, n);
    bincount_finalize<<<(NBINS + 255) / 256, 256, 0, stream>>>(ws, out);
}